// DglGCNNet_72224170049984
// MI455X (gfx1250) — compile-verified
//
#include <hip/hip_runtime.h>
#include <hip/hip_bf16.h>
#include <stdint.h>

#define D_IN  128
#define D_HID 128

typedef __attribute__((ext_vector_type(2))) float v2f;
typedef __attribute__((ext_vector_type(8))) float v8f;

// ---------------------------------------------------------------- utilities

__device__ __forceinline__ void atomAddF(float* p, float v) {
#if defined(__HIP_DEVICE_COMPILE__)
    unsafeAtomicAdd(p, v);   // hardware global_atomic_add_f32 (no CAS loop)
#else
    atomicAdd(p, v);
#endif
}

__global__ void zero_f32_kernel(float* __restrict__ p, long long n) {
    long long i = (long long)blockIdx.x * blockDim.x + threadIdx.x;
    if (i < n) p[i] = 0.0f;
}

__global__ void degree_kernel(const long long* __restrict__ src,
                              const long long* __restrict__ dst,
                              float* __restrict__ deg_s,
                              float* __restrict__ deg_d,
                              long long nE) {
    long long e = (long long)blockIdx.x * blockDim.x + threadIdx.x;
    if (e < nE) {
        atomAddF(&deg_s[(int)src[e]], 1.0f);
        atomAddF(&deg_d[(int)dst[e]], 1.0f);
    }
}

__global__ void norm_kernel(float* __restrict__ a, float* __restrict__ b, int n) {
    int i = blockIdx.x * blockDim.x + threadIdx.x;
    if (i < n) {
        float va = a[i]; a[i] = (va > 0.0f) ? rsqrtf(va) : 0.0f;
        float vb = b[i]; b[i] = (vb > 0.0f) ? rsqrtf(vb) : 0.0f;
    }
}

// -------------------------------------------------- fused GEMM + src-norm
// H[n, DOUT] = (X[n, 128] @ W[128, DOUT]) * norm_src[n]
// One wave -> 16 rows x all DOUT columns via V_WMMA_F32_16X16X4_F32.
// Block = 256 threads = 8 waves = 128 rows.
//
// W is staged in LDS *pair-interleaved*:
//   Ws[(k>>1)*2*DOUT + col*2 + (k&1)] = W[k][col]
// so each B fragment (W[k+kk][col], W[k+kk+1][col]) is one contiguous,
// 8B-aligned ds_load_b64 straight into an even VGPR pair (no shuffle movs).
//
// A loads are unconditional global_load_b64 from a clamped row pointer;
// out-of-range tail rows are zeroed with a mask multiply (keeps EXEC full
// and avoids exec-mask branches around loads).
template <int DOUT>
__global__ __launch_bounds__(256)
void gemm_norm_kernel(const float* __restrict__ X,
                      const float* __restrict__ W,
                      const float* __restrict__ norm_src,
                      float* __restrict__ H,
                      int n_rows) {
    constexpr int CT = DOUT / 16;
    __shared__ float Ws[D_IN * DOUT];            // 64KB (DOUT=128) <= 320KB/WGP

    for (int i = threadIdx.x; i < D_IN * DOUT; i += 256) {
        const int k   = i / DOUT;
        const int col = i - k * DOUT;
        Ws[(k >> 1) * (2 * DOUT) + col * 2 + (k & 1)] = W[i];
    }
    __syncthreads();

    const int wave = threadIdx.x >> 5;
    const int lane = threadIdx.x & 31;
    const int half = lane >> 4;                  // 0: K={0,1}, 1: K={2,3}
    const int l16  = lane & 15;
    const int kk   = half * 2;
    const int row_base = blockIdx.x * 128 + wave * 16;

    const int   arow   = row_base + l16;
    const float amsk   = (arow < n_rows) ? 1.0f : 0.0f;
    const int   arow_c = (arow < n_rows) ? arow : (n_rows - 1);
    const float* __restrict__ xrow = X + (long long)arow_c * D_IN + kk;

    v8f acc[CT];
#pragma unroll
    for (int c = 0; c < CT; ++c) acc[c] = (v8f)0.0f;

#pragma unroll 2
    for (int k = 0; k < D_IN; k += 4) {
        v2f a = *reinterpret_cast<const v2f*>(xrow + k);   // global_load_b64
        a = a * amsk;                                      // zero tail rows
        const float* __restrict__ wsrow = &Ws[((k + kk) >> 1) * (2 * DOUT)];
#pragma unroll
        for (int c = 0; c < CT; ++c) {
            v2f b = *reinterpret_cast<const v2f*>(wsrow + (c * 16 + l16) * 2); // ds_load_b64
            acc[c] = __builtin_amdgcn_wmma_f32_16x16x4_f32(
                false, a, false, b, (short)0, acc[c], false, false);
        }
    }

#pragma unroll
    for (int r = 0; r < 8; ++r) {
        const int m    = half ? (r + 8) : r;
        const int grow = row_base + m;
        if (grow < n_rows) {
            const float ns = norm_src[grow];
#pragma unroll
            for (int c = 0; c < CT; ++c)
                H[(long long)grow * DOUT + c * 16 + l16] = acc[c][r] * ns;
        }
    }
}

// ------------------------------------------------ edge gather + scatter-add
// F/4 lanes per edge; float4 gather from h[src], 4x f32 atomics into agg[dst].
template <int F>
__global__ __launch_bounds__(256)
void scatter_add_kernel(const float* __restrict__ H,
                        const long long* __restrict__ src,
                        const long long* __restrict__ dst,
                        float* __restrict__ AGG,
                        long long nE) {
    constexpr int CPE = F / 4;
    long long idx = (long long)blockIdx.x * blockDim.x + threadIdx.x;
    long long e = idx / CPE;
    if (e >= nE) return;
    const int c = (int)(idx % CPE);
    const int s = (int)src[e];
    const int d = (int)dst[e];
    const float4 v = *reinterpret_cast<const float4*>(H + (long long)s * F + c * 4);
    float* p = AGG + (long long)d * F + c * 4;
    atomAddF(p + 0, v.x);
    atomAddF(p + 1, v.y);
    atomAddF(p + 2, v.z);
    atomAddF(p + 3, v.w);
}

// ------------------------------------------------ dst-norm + bias (+ tanh)
template <int F, bool ACT>
__global__ void finalize_kernel(float* __restrict__ AGG,
                                const float* __restrict__ norm_dst,
                                const float* __restrict__ bias,
                                int n_rows) {
    long long idx = (long long)blockIdx.x * blockDim.x + threadIdx.x;
    long long total = (long long)n_rows * F;
    if (idx >= total) return;
    const int node = (int)(idx / F);
    const int f    = (int)(idx % F);
    float v = AGG[idx] * norm_dst[node] + bias[f];
    if (ACT) v = tanhf(v);
    AGG[idx] = v;
}

// ------------------------------------------------------------------ driver

static inline unsigned cdivu(long long a, long long b) {
    return (unsigned)((a + b - 1) / b);
}

extern "C" void kernel_launch(void* const* d_in, const int* in_sizes, int n_in,
                              void* d_out, int out_size, void* d_ws, size_t ws_size,
                              hipStream_t stream) {
    const float*     features = (const float*)d_in[0];
    const long long* edge     = (const long long*)d_in[1];   // int64 per reference
    const float* W0 = (const float*)d_in[2];
    const float* b0 = (const float*)d_in[3];
    const float* W1 = (const float*)d_in[4];
    const float* b1 = (const float*)d_in[5];
    const float* W2 = (const float*)d_in[6];
    const float* b2 = (const float*)d_in[7];
    const float* W3 = (const float*)d_in[8];
    const float* b3 = (const float*)d_in[9];

    const int       N = in_sizes[0] / D_IN;
    const long long E = (long long)in_sizes[1] / 2;
    const long long* src = edge;
    const long long* dst = edge + E;

    char* ws = (char*)d_ws;
    float* H  = (float*)ws; ws += (size_t)N * D_HID * sizeof(float);
    float* P  = (float*)ws; ws += (size_t)N * D_HID * sizeof(float);
    float* ns = (float*)ws; ws += (size_t)N * sizeof(float);
    float* nd = (float*)ws; ws += (size_t)N * sizeof(float);
    float* out = (float*)d_out;

    const unsigned gemm_blocks = cdivu(N, 128);
    const long long NF128 = (long long)N * 128;
    const long long NF64  = (long long)N * 64;

    // ---- degree norms (ns, nd contiguous: one zero launch) ----
    zero_f32_kernel<<<cdivu(2LL * N, 256), 256, 0, stream>>>(ns, 2LL * N);
    degree_kernel<<<cdivu(E, 256), 256, 0, stream>>>(src, dst, ns, nd, E);
    norm_kernel<<<cdivu(N, 256), 256, 0, stream>>>(ns, nd, N);

    // ---- layer 0: features -> P ----
    gemm_norm_kernel<128><<<gemm_blocks, 256, 0, stream>>>(features, W0, ns, H, N);
    zero_f32_kernel<<<cdivu(NF128, 256), 256, 0, stream>>>(P, NF128);
    scatter_add_kernel<128><<<cdivu(E * 32, 256), 256, 0, stream>>>(H, src, dst, P, E);
    finalize_kernel<128, true><<<cdivu(NF128, 256), 256, 0, stream>>>(P, nd, b0, N);

    // ---- layer 1: P -> P (stream-ordered: gemm drains P into H first) ----
    gemm_norm_kernel<128><<<gemm_blocks, 256, 0, stream>>>(P, W1, ns, H, N);
    zero_f32_kernel<<<cdivu(NF128, 256), 256, 0, stream>>>(P, NF128);
    scatter_add_kernel<128><<<cdivu(E * 32, 256), 256, 0, stream>>>(H, src, dst, P, E);
    finalize_kernel<128, true><<<cdivu(NF128, 256), 256, 0, stream>>>(P, nd, b1, N);

    // ---- layer 2: P -> P ----
    gemm_norm_kernel<128><<<gemm_blocks, 256, 0, stream>>>(P, W2, ns, H, N);
    zero_f32_kernel<<<cdivu(NF128, 256), 256, 0, stream>>>(P, NF128);
    scatter_add_kernel<128><<<cdivu(E * 32, 256), 256, 0, stream>>>(H, src, dst, P, E);
    finalize_kernel<128, true><<<cdivu(NF128, 256), 256, 0, stream>>>(P, nd, b2, N);

    // ---- layer 3 (D_OUT=64, no tanh): P -> d_out ----
    gemm_norm_kernel<64><<<gemm_blocks, 256, 0, stream>>>(P, W3, ns, H, N);
    zero_f32_kernel<<<cdivu(NF64, 256), 256, 0, stream>>>(out, NF64);
    scatter_add_kernel<64><<<cdivu(E * 16, 256), 256, 0, stream>>>(H, src, dst, out, E);
    finalize_kernel<64, false><<<cdivu(NF64, 256), 256, 0, stream>>>(out, nd, b3, N);
}